// ModelNew_57415122813199
// MI455X (gfx1250) — compile-verified
//
#include <hip/hip_runtime.h>

typedef __attribute__((ext_vector_type(2))) float v2f;
typedef __attribute__((ext_vector_type(8))) float v8f;

#define CIN 8
#define COUT 64
#define KDIM 72          // 8*3*3
#define IH 128
#define IW 128
#define OHH 126
#define OWW 126
#define XT_ROWS 6        // 4 output rows + 2 halo
#define XT_COLS 34       // 32 output cols + 2 halo
#define NPOS 128         // 4 rows * 32 cols per block tile
#define NGROUPS 16
#define GCOUNT 63504.0f  // 4 channels * 126 * 126
#define NBINS 2048       // 128 batches * 16 groups

// ---------------------------------------------------------------------------
// Stage weights as k-pair-interleaved [k>>1][n*2 + (k&1)] and load the raw
// input halo tile.
// ---------------------------------------------------------------------------
__device__ __forceinline__ void load_raw(const float* __restrict__ x,
                                         const float* __restrict__ cw,
                                         float* __restrict__ wlds,
                                         float* __restrict__ xraw,
                                         int b, int oh0, int ow0, int tid) {
  // wlds[(k>>1)*128 + n*2 + (k&1)] = conv_weight[n][k]
  for (int idx = tid; idx < KDIM * COUT; idx += 256) {
    int kp = idx >> 7;           // 0..35
    int rem = idx & 127;
    int n = rem >> 1;
    int k = kp * 2 + (rem & 1);
    wlds[idx] = cw[n * KDIM + k];
  }
  // xraw[(cin*6 + r)*34 + c] = x[b][cin][oh0+r][ow0+c] (0 padded)
  for (int idx = tid; idx < CIN * XT_ROWS * XT_COLS; idx += 256) {
    int cin = idx / (XT_ROWS * XT_COLS);
    int rem = idx % (XT_ROWS * XT_COLS);
    int rr = rem / XT_COLS;
    int cc = rem % XT_COLS;
    int ih = oh0 + rr;
    int iw = ow0 + cc;
    float v = 0.0f;
    if (ih < IH && iw < IW) v = x[((b * CIN + cin) * IH + ih) * IW + iw];
    xraw[idx] = v;
  }
}

// ---------------------------------------------------------------------------
// Expand raw halo tile -> k-pair-interleaved im2col tile:
//   xcol[(k>>1)*256 + p*2 + (k&1)] = xraw[(cin*6 + r + kh)*34 + c + kw]
// so the WMMA loop's A-pair (k even, k+1) is one aligned ds_load_b64 with a
// step-affine immediate offset.
// ---------------------------------------------------------------------------
__device__ __forceinline__ void expand_im2col(const float* __restrict__ xraw,
                                              float* __restrict__ xcol,
                                              int tid) {
  for (int idx = tid; idx < KDIM * NPOS; idx += 256) {
    int kp = idx >> 8;           // 0..35
    int rem = idx & 255;
    int p = rem >> 1;
    int k = kp * 2 + (rem & 1);
    int cin = k / 9;
    int r9 = k % 9;
    int r = p >> 5;
    int c = p & 31;
    xcol[idx] = xraw[(cin * XT_ROWS + r + r9 / 3) * XT_COLS + c + (r9 % 3)];
  }
}

// ---------------------------------------------------------------------------
// Per-wave implicit-GEMM conv tile: 16 spatial positions x 64 channels,
// K = 72 via 18 steps of V_WMMA_F32_16X16X4_F32, 4 independent accumulators.
// All LDS addresses are base + immediate (step/nt) offsets.
// ---------------------------------------------------------------------------
__device__ __forceinline__ void conv_tile(const float* __restrict__ wlds,
                                          const float* __restrict__ xcol,
                                          int wave, int lane, v8f acc[4]) {
  const int half = lane >> 4;      // selects K pair (0,1) vs (2,3)
  const int l15 = lane & 15;
  const int p = wave * 16 + l15;   // A-matrix row = spatial position

  const float* abase = xcol + half * 256 + p * 2;    // +step*512 floats
  const float* bbase = wlds + half * 128 + l15 * 2;  // +step*256 + nt*32

  acc[0] = (v8f){};
  acc[1] = (v8f){};
  acc[2] = (v8f){};
  acc[3] = (v8f){};

#pragma unroll
  for (int step = 0; step < 18; ++step) {
    v2f a = *(const v2f*)(abase + step * 512);
#pragma unroll
    for (int nt = 0; nt < 4; ++nt) {
      v2f bf = *(const v2f*)(bbase + step * 256 + nt * 32);
      acc[nt] = __builtin_amdgcn_wmma_f32_16x16x4_f32(
          false, a, false, bf, (short)0, acc[nt], false, false);
    }
  }
}

// ---------------------------------------------------------------------------
// Pass 1: conv + bias, masked per-(batch,group) sum / sumsq into d_ws bins.
// Grid (4 col-chunks, 32 row-groups, 128 batches), 256 threads.
// ---------------------------------------------------------------------------
__global__ __launch_bounds__(256) void k_conv_stats(
    const float* __restrict__ x, const float* __restrict__ cw,
    const float* __restrict__ cb, float* __restrict__ stats) {
  __shared__ float wlds[KDIM * 2 * COUT / 2 + 0];  // 4608
  __shared__ float xraw[CIN * XT_ROWS * XT_COLS];
  __shared__ float xcol[KDIM * NPOS];
  __shared__ float gbin[NGROUPS * 2];

  const int tid = threadIdx.x;
  const int b = blockIdx.z;
  const int rg = blockIdx.y;
  const int chunk = blockIdx.x;
  const int oh0 = rg * 4;
  const int ow0 = chunk * 32;

  if (tid < NGROUPS * 2) gbin[tid] = 0.0f;
  load_raw(x, cw, wlds, xraw, b, oh0, ow0, tid);
  __syncthreads();
  expand_im2col(xraw, xcol, tid);
  __syncthreads();

  const int wave = tid >> 5;
  const int lane = tid & 31;
  v8f acc[4];
  conv_tile(wlds, xcol, wave, lane, acc);

  const int l15 = lane & 15;
  const int pbase = wave * 16 + ((lane >> 4) << 3);  // D: lanes>=16 -> M+8

#pragma unroll
  for (int nt = 0; nt < 4; ++nt) {
    const int n = nt * 16 + l15;
    const float bias = cb[n];
    float s = 0.0f, s2 = 0.0f;
#pragma unroll
    for (int i = 0; i < 8; ++i) {
      const int p = pbase + i;
      const int oh = oh0 + (p >> 5);
      const int ow = ow0 + (p & 31);
      float v = acc[nt][i] + bias;
      v = (oh < OHH && ow < OWW) ? v : 0.0f;
      s += v;
      s2 += v * v;
    }
    // lanes {l, l^16, l^1, l^2} all carry channels of the same 4-ch group
    s += __shfl_xor(s, 16); s2 += __shfl_xor(s2, 16);
    s += __shfl_xor(s, 1);  s2 += __shfl_xor(s2, 1);
    s += __shfl_xor(s, 2);  s2 += __shfl_xor(s2, 2);
    if (lane < 16 && (lane & 3) == 0) {
      const int g = nt * 4 + (lane >> 2);
      atomicAdd(&gbin[g * 2 + 0], s);
      atomicAdd(&gbin[g * 2 + 1], s2);
    }
  }
  __syncthreads();
  if (tid < NGROUPS) {
    atomicAdd(&stats[b * NGROUPS + tid], gbin[tid * 2 + 0]);
    atomicAdd(&stats[NBINS + b * NGROUPS + tid], gbin[tid * 2 + 1]);
  }
}

// ---------------------------------------------------------------------------
// Pass 2: fold GroupNorm stats + gn affine + per-channel scale into per-(b,c)
// alpha/beta so the output pass is a single FMA per element.
// ---------------------------------------------------------------------------
__global__ __launch_bounds__(256) void k_finalize(
    const float* __restrict__ stats, const float* __restrict__ gnw,
    const float* __restrict__ gnb, const float* __restrict__ scl,
    float* __restrict__ alpha, float* __restrict__ beta) {
  const int t = blockIdx.x * blockDim.x + threadIdx.x;
  if (t >= 128 * COUT) return;
  const int b = t >> 6;
  const int c = t & 63;
  const int g = c >> 2;
  const float sum = stats[b * NGROUPS + g];
  const float sq = stats[NBINS + b * NGROUPS + g];
  const float mean = sum * (1.0f / GCOUNT);
  const float var = sq * (1.0f / GCOUNT) - mean * mean;
  const float rstd = rsqrtf(var + 1e-5f);
  const float aw = rstd * gnw[c] * scl[c];
  alpha[t] = aw;
  beta[t] = gnb[c] * scl[c] - mean * aw;
}

// ---------------------------------------------------------------------------
// Pass 3: recompute conv tile, apply alpha/beta, 4x4 maxpool + clamp.
// Grid (4 col-chunks, 31 pool rows, 128 batches), 256 threads.
// ---------------------------------------------------------------------------
__global__ __launch_bounds__(256) void k_conv_pool_out(
    const float* __restrict__ x, const float* __restrict__ cw,
    const float* __restrict__ cb, const float* __restrict__ alpha,
    const float* __restrict__ beta, float* __restrict__ out) {
  __shared__ float wlds[KDIM * COUT];
  __shared__ float xraw[CIN * XT_ROWS * XT_COLS];
  __shared__ float xcol[KDIM * NPOS];
  __shared__ float olds[NPOS * COUT];  // [4 rows * 32 cols][64 channels]

  const int tid = threadIdx.x;
  const int b = blockIdx.z;
  const int rg = blockIdx.y;   // pool row 0..30
  const int chunk = blockIdx.x;
  const int oh0 = rg * 4;
  const int ow0 = chunk * 32;

  load_raw(x, cw, wlds, xraw, b, oh0, ow0, tid);
  __syncthreads();
  expand_im2col(xraw, xcol, tid);
  __syncthreads();

  const int wave = tid >> 5;
  const int lane = tid & 31;
  v8f acc[4];
  conv_tile(wlds, xcol, wave, lane, acc);

  const int l15 = lane & 15;
  const int pbase = wave * 16 + ((lane >> 4) << 3);

#pragma unroll
  for (int nt = 0; nt < 4; ++nt) {
    const int n = nt * 16 + l15;
    const float bias = cb[n];
    const float av = alpha[b * COUT + n];
    const float bv = beta[b * COUT + n];
#pragma unroll
    for (int i = 0; i < 8; ++i) {
      const int p = pbase + i;
      olds[p * COUT + n] = (acc[nt][i] + bias) * av + bv;
    }
  }
  __syncthreads();

  // 8 pool cols * 64 channels = 512 outputs; 2 per thread
#pragma unroll
  for (int j = 0; j < 2; ++j) {
    const int o = tid + j * 256;
    const int pcl = o >> 6;   // local pool col 0..7
    const int n = o & 63;
    float m = -3.4e38f;
#pragma unroll
    for (int r = 0; r < 4; ++r)
#pragma unroll
      for (int dc = 0; dc < 4; ++dc)
        m = fmaxf(m, olds[((r << 5) + pcl * 4 + dc) * COUT + n]);
    m = fminf(fmaxf(m, 0.0f), 1.0f);
    const int pc = chunk * 8 + pcl;
    if (pc < 31) out[(b * COUT + n) * 961 + rg * 31 + pc] = m;
  }
}

// ---------------------------------------------------------------------------
extern "C" void kernel_launch(void* const* d_in, const int* in_sizes, int n_in,
                              void* d_out, int out_size, void* d_ws,
                              size_t ws_size, hipStream_t stream) {
  const float* x   = (const float*)d_in[0];
  const float* cw  = (const float*)d_in[1];
  const float* cb  = (const float*)d_in[2];
  const float* gnw = (const float*)d_in[3];
  const float* gnb = (const float*)d_in[4];
  const float* scl = (const float*)d_in[5];
  float* out = (float*)d_out;

  float* stats = (float*)d_ws;               // 2*2048 floats (sum, sumsq)
  float* alpha = stats + 2 * NBINS;          // 8192 floats
  float* beta  = alpha + 128 * COUT;         // 8192 floats

  hipMemsetAsync(stats, 0, 2 * NBINS * sizeof(float), stream);

  k_conv_stats<<<dim3(4, 32, 128), 256, 0, stream>>>(x, cw, cb, stats);
  k_finalize<<<dim3(32), 256, 0, stream>>>(stats, gnw, gnb, scl, alpha, beta);
  k_conv_pool_out<<<dim3(4, 31, 128), 256, 0, stream>>>(x, cw, cb, alpha, beta,
                                                        out);
}